// Generator_80633716015375
// MI455X (gfx1250) — compile-verified
//
#include <hip/hip_runtime.h>

#define TT 48
#define NA 35
#define NM 18
#define NN 53
#define HH 32

typedef float v2f __attribute__((ext_vector_type(2)));
typedef float v8f __attribute__((ext_vector_type(8)));

__device__ __forceinline__ float wsum(float v) {
#pragma unroll
    for (int o = 16; o >= 1; o >>= 1) v += __shfl_xor(v, o, 32);
    return v;
}
__device__ __forceinline__ float wmax(float v) {
#pragma unroll
    for (int o = 16; o >= 1; o >>= 1) v = fmaxf(v, __shfl_xor(v, o, 32));
    return v;
}
__device__ __forceinline__ float sigm(float x) { return 1.0f / (1.0f + __expf(-x)); }
__device__ __forceinline__ float tanh_f(float x) { return 1.0f - 2.0f / (__expf(2.0f * x) + 1.0f); }

__device__ __forceinline__ v8f wmma4(v2f a, v2f b, v8f c) {
    // D = A(16x4) * B(4x16) + C, fp32
    return __builtin_amdgcn_wmma_f32_16x16x4_f32(false, a, false, b, (short)0, c, false, false);
}

// One GRU block update: x[nrows,32] @ wih^T + h[nrows,32] @ whh^T -> gates -> h
static __device__ void gru_block(const float* xbuf, float* hbuf,
                                 int nrows, int mtiles,
                                 const float* __restrict__ gwih,
                                 const float* __restrict__ gwhh,
                                 const float* sbih, const float* sbhh, int lane) {
    const int half = lane >> 4;
    const int l16  = lane & 15;
    const v8f vz = {0.f, 0.f, 0.f, 0.f, 0.f, 0.f, 0.f, 0.f};
    for (int mt = 0; mt < mtiles; ++mt) {
        v8f ai[6], ah[6];
#pragma unroll
        for (int n = 0; n < 6; ++n) { ai[n] = vz; ah[n] = vz; }
        const int row = mt * 16 + l16;
        const bool rv = row < nrows;
        const int rr = rv ? row : 0;
#pragma unroll
        for (int kk = 0; kk < 8; ++kk) {
            const int k0 = kk * 4 + half * 2;
            v2f ax, ahf;
            ax.x  = xbuf[rr * HH + k0];  ax.y  = xbuf[rr * HH + k0 + 1];
            ahf.x = hbuf[rr * HH + k0];  ahf.y = hbuf[rr * HH + k0 + 1];
            if (!rv) { ax.x = 0.f; ax.y = 0.f; ahf.x = 0.f; ahf.y = 0.f; }
            const v2f* wiv = (const v2f*)gwih;   // wih[n][k] row-major -> B[k][c]=wih[c][k]
            const v2f* whv = (const v2f*)gwhh;
#pragma unroll
            for (int n = 0; n < 6; ++n) {
                const int c = n * 16 + l16;
                v2f bi = wiv[c * 16 + (k0 >> 1)];
                v2f bh = whv[c * 16 + (k0 >> 1)];
                ai[n] = wmma4(ax, bi, ai[n]);
                ah[n] = wmma4(ahf, bh, ah[n]);
            }
        }
        // gate math: fragments (0,2,4) carry h-cols l16, (1,3,5) carry 16+l16
#pragma unroll
        for (int v = 0; v < 8; ++v) {
            const int r = mt * 16 + v + half * 8;
            if (r < nrows) {
#pragma unroll
                for (int part = 0; part < 2; ++part) {
                    const int hc = part * 16 + l16;
                    float ir  = ai[0 + part][v] + sbih[hc];
                    float hr  = ah[0 + part][v] + sbhh[hc];
                    float iz  = ai[2 + part][v] + sbih[32 + hc];
                    float hz  = ah[2 + part][v] + sbhh[32 + hc];
                    float in_ = ai[4 + part][v] + sbih[64 + hc];
                    float hn  = ah[4 + part][v] + sbhh[64 + hc];
                    float rg = sigm(ir + hr);
                    float zg = sigm(iz + hz);
                    float ng = tanh_f(in_ + rg * hn);
                    float ho = hbuf[r * HH + hc];
                    hbuf[r * HH + hc] = (1.0f - zg) * ng + zg * ho;
                }
            }
        }
    }
}

__global__ __launch_bounds__(32) void chgat_gru_kernel(
    const float* __restrict__ g_Xa, const float* __restrict__ g_Xm,
    const float* __restrict__ g_ctx, const float* __restrict__ g_adj,
    const float* __restrict__ g_adjn,
    const float* __restrict__ e_aid, const float* __restrict__ e_amo,
    const float* __restrict__ e_awd, const float* __restrict__ e_ahr,
    const float* __restrict__ e_mwi, const float* __restrict__ e_mid,
    const float* __restrict__ e_mmo, const float* __restrict__ e_mwd,
    const float* __restrict__ e_mhr,
    const float* __restrict__ g_Wxa, const float* __restrict__ g_Wxm,
    const float* __restrict__ g_Wua, const float* __restrict__ g_Wum,
    const float* __restrict__ a_aa, const float* __restrict__ a_am,
    const float* __restrict__ a_ma, const float* __restrict__ a_mm,
    const float* __restrict__ g_wihA, const float* __restrict__ g_whhA,
    const float* __restrict__ g_bihA, const float* __restrict__ g_bhhA,
    const float* __restrict__ g_wihM, const float* __restrict__ g_whhM,
    const float* __restrict__ g_bihM, const float* __restrict__ g_bhhM,
    const int* __restrict__ g_Xae, const int* __restrict__ g_Xme,
    float* __restrict__ d_out, int nb) {

    __shared__ float s_att[NN * 56];     // att rows (cols 53..55 stay 0)
    __shared__ float s_attri[56 * HH];   // value matrix, K-padded (rows 53..55 zero)
    __shared__ float s_xA[NA * HH];
    __shared__ float s_xM[NM * HH];
    __shared__ float s_hA[NA * HH];
    __shared__ float s_hM[NM * HH];
    __shared__ float s_in14[NN * 14];
    __shared__ float s_xrA[NA * 6];
    __shared__ float s_xrM[NM * 4];
    __shared__ float s_sd[4 * NN];       // srcColA, srcColM, dstRowA, dstRowM
    __shared__ float s_ctxc[4 * NN];
    __shared__ float s_aH[8 * HH];       // a-vec hidden coefs
    __shared__ float s_alast[4];
    __shared__ float s_bihA[96], s_bhhA[96], s_bihM[96], s_bhhM[96];
    __shared__ float s_emb[150 * 2];
    __shared__ float s_Wua[14 * HH], s_Wum[14 * HH], s_Wxa[6 * HH], s_Wxm[4 * HH];
    __shared__ float s_adjn[NN * NN];
    __shared__ unsigned s_mask[NN * 2];

    const int lane = threadIdx.x;
    const int b = blockIdx.x;
    if (b >= nb) return;
    const int half = lane >> 4;
    const int l16  = lane & 15;

    // ---------------- one-time init ----------------
    for (int i = lane; i < 14 * HH; i += 32) { s_Wua[i] = g_Wua[i]; s_Wum[i] = g_Wum[i]; }
    for (int i = lane; i < 6 * HH; i += 32) s_Wxa[i] = g_Wxa[i];
    for (int i = lane; i < 4 * HH; i += 32) s_Wxm[i] = g_Wxm[i];
    for (int i = lane; i < 96; i += 32) {
        s_bihA[i] = g_bihA[i]; s_bhhA[i] = g_bhhA[i];
        s_bihM[i] = g_bihM[i]; s_bhhM[i] = g_bhhM[i];
    }
    // embeddings packed: offsets {0,35,48,55,79,88,106,119,126}, 150 rows of 2
    {
        const float* tabs[9] = {e_aid, e_amo, e_awd, e_ahr, e_mwi, e_mid, e_mmo, e_mwd, e_mhr};
        const int rows[9] = {35, 13, 7, 24, 9, 18, 13, 7, 24};
        int off = 0;
        for (int tb = 0; tb < 9; ++tb) {
            for (int i = lane; i < rows[tb] * 2; i += 32) s_emb[off * 2 + i] = tabs[tb][i];
            off += rows[tb];
        }
    }
    {
        const float* av[4] = {a_aa, a_am, a_ma, a_mm};
#pragma unroll
        for (int m = 0; m < 4; ++m) {
            s_aH[m * HH + lane]       = av[m][lane];        // src hidden coefs
            s_aH[(4 + m) * HH + lane] = av[m][92 + lane];   // dst hidden coefs
        }
        if (lane == 0) {
#pragma unroll
            for (int m = 0; m < 4; ++m) s_alast[m] = av[m][184];
        }
    }
    // context constants per node (batch/time invariant)
    for (int i = lane; i < NN; i += 32) {
        const float* cx = g_ctx + i * 60;
        const float* aS0 = (i < NA) ? a_aa : a_ma;
        const float* aS1 = (i < NA) ? a_am : a_mm;
        const float* aD0 = (i < NA) ? a_aa : a_am;
        const float* aD1 = (i < NA) ? a_ma : a_mm;
        float t0 = 0.f, t1 = 0.f, t2 = 0.f, t3 = 0.f;
        for (int c = 0; c < 60; ++c) {
            float cv = cx[c];
            t0 += cv * aS0[32 + c];  t1 += cv * aS1[32 + c];
            t2 += cv * aD0[124 + c]; t3 += cv * aD1[124 + c];
        }
        s_ctxc[0 * NN + i] = t0; s_ctxc[1 * NN + i] = t1;
        s_ctxc[2 * NN + i] = t2; s_ctxc[3 * NN + i] = t3;
    }
    for (int i = lane; i < NN * NN; i += 32) s_adjn[i] = g_adjn[i];
    for (int i = lane; i < NN; i += 32) {
        unsigned w0 = 0, w1 = 0;
        for (int j = 0; j < NN; ++j) {
            if (g_adj[i * NN + j] > 0.0f) {
                if (j < 32) w0 |= (1u << j); else w1 |= (1u << (j - 32));
            }
        }
        s_mask[i * 2] = w0; s_mask[i * 2 + 1] = w1;
    }
    for (int i = lane; i < NN * 56; i += 32) s_att[i] = 0.f;
    for (int i = lane; i < 3 * HH; i += 32) s_attri[53 * HH + i] = 0.f;
    for (int i = lane; i < NA * HH; i += 32) s_hA[i] = 0.f;
    for (int i = lane; i < NM * HH; i += 32) s_hM[i] = 0.f;
    __syncthreads();

    // ---------------- sequential scan over T ----------------
#pragma unroll 1
    for (int t = 0; t < TT; ++t) {
        const float* xa = g_Xa + ((size_t)b * TT + t) * (NA * 6);
        const float* xm = g_Xm + ((size_t)b * TT + t) * (NM * 4);
        const int* exa = g_Xae + ((size_t)b * TT + t) * (NA * 4);
        const int* exm = g_Xme + ((size_t)b * TT + t) * (NM * 5);
        for (int i = lane; i < NA * 6; i += 32) s_xrA[i] = xa[i];
        for (int i = lane; i < NM * 4; i += 32) s_xrM[i] = xm[i];
        if (t + 1 < TT) {  // prefetch next step's inputs
            __builtin_prefetch(xa + NA * 6 + lane * 16, 0, 0);
            __builtin_prefetch(xm + NM * 4 + lane * 8, 0, 0);
        }

        // gather embeddings -> in14
        for (int i = lane; i < NN; i += 32) {
            if (i < NA) {
#pragma unroll
                for (int k = 0; k < 6; ++k) s_in14[i * 14 + k] = s_xrA[i * 6 + k];
                const int offs[4] = {0, 35, 48, 55};
#pragma unroll
                for (int e = 0; e < 4; ++e) {
                    int r = exa[i * 4 + e];
                    int bse = (offs[e] + r) * 2;
                    s_in14[i * 14 + 6 + 2 * e]     = s_emb[bse];
                    s_in14[i * 14 + 6 + 2 * e + 1] = s_emb[bse + 1];
                }
            } else {
                int j = i - NA;
#pragma unroll
                for (int k = 0; k < 4; ++k) s_in14[i * 14 + k] = s_xrM[j * 4 + k];
                const int offs[5] = {79, 88, 106, 119, 126};
#pragma unroll
                for (int e = 0; e < 5; ++e) {
                    int r = exm[j * 5 + e];
                    int bse = (offs[e] + r) * 2;
                    s_in14[i * 14 + 4 + 2 * e]     = s_emb[bse];
                    s_in14[i * 14 + 4 + 2 * e + 1] = s_emb[bse + 1];
                }
            }
        }

        // attri (aggregation values): raw features only
#pragma unroll 1
        for (int i = 0; i < NA; ++i) {
            float s = 0.f;
#pragma unroll
            for (int k = 0; k < 6; ++k) s += s_xrA[i * 6 + k] * s_Wxa[k * HH + lane];
            s_attri[i * HH + lane] = s;
        }
#pragma unroll 1
        for (int j = 0; j < NM; ++j) {
            float s = 0.f;
#pragma unroll
            for (int k = 0; k < 4; ++k) s += s_xrM[j * 4 + k] * s_Wxm[k * HH + lane];
            s_attri[(NA + j) * HH + lane] = s;
        }

        // attention hidden features -> per-node src/dst scalars
#pragma unroll 1
        for (int i = 0; i < NN; ++i) {
            const float* Wu = (i < NA) ? s_Wua : s_Wum;
            float fh = 0.f;
#pragma unroll
            for (int k = 0; k < 14; ++k) fh += s_in14[i * 14 + k] * Wu[k * HH + lane];
            const int blk = (i < NA) ? 0 : 1;
            float v0 = fh * s_aH[(blk ? 2 : 0) * HH + lane];
            float v1 = fh * s_aH[(blk ? 3 : 1) * HH + lane];
            float v2 = fh * s_aH[(blk ? 5 : 4) * HH + lane];
            float v3 = fh * s_aH[(blk ? 7 : 6) * HH + lane];
            v0 = wsum(v0); v1 = wsum(v1); v2 = wsum(v2); v3 = wsum(v3);
            if (lane == 0) {
                s_sd[0 * NN + i] = v0 + s_ctxc[0 * NN + i];
                s_sd[1 * NN + i] = v1 + s_ctxc[1 * NN + i];
                s_sd[2 * NN + i] = v2 + s_ctxc[2 * NN + i];
                s_sd[3 * NN + i] = v3 + s_ctxc[3 * NN + i];
            }
        }

        // e + masked softmax rows
#pragma unroll 1
        for (int i = 0; i < NN; ++i) {
            const int Rm = (i < NA) ? 0 : 1;
            const float sA = s_sd[0 * NN + i], sM = s_sd[1 * NN + i];
            const float* dstArr = (Rm == 0) ? (s_sd + 2 * NN) : (s_sd + 3 * NN);
            const float al0 = s_alast[Rm * 2 + 0];
            const float al1 = s_alast[Rm * 2 + 1];
            const unsigned m0 = s_mask[i * 2], m1 = s_mask[i * 2 + 1];
            const int j1 = lane;
            float e1 = ((j1 < NA) ? sA : sM) + dstArr[j1] +
                       s_adjn[i * NN + j1] * ((j1 < NA) ? al0 : al1);
            e1 = (e1 >= 0.f) ? e1 : 0.2f * e1;
            float val1 = ((m0 >> j1) & 1u) ? e1 : -1.0e12f;
            const int j2 = lane + 32;
            const bool has2 = (j2 < NN);
            float val2 = -3.0e38f;
            if (has2) {
                float e2 = ((j2 < NA) ? sA : sM) + dstArr[j2] +
                           s_adjn[i * NN + j2] * ((j2 < NA) ? al0 : al1);
                e2 = (e2 >= 0.f) ? e2 : 0.2f * e2;
                val2 = ((m1 >> (j2 - 32)) & 1u) ? e2 : -1.0e12f;
            }
            const float mx = wmax(fmaxf(val1, val2));
            const float p1 = __expf(val1 - mx);
            const float p2 = has2 ? __expf(val2 - mx) : 0.f;
            const float inv = 1.0f / wsum(p1 + p2);
            s_att[i * 56 + j1] = p1 * inv;
            if (has2) s_att[i * 56 + j2] = p2 * inv;
        }

        // out = att @ attri  -> split into xA / xM   (WMMA fp32 16x16x4)
        {
            const v8f vz = {0.f, 0.f, 0.f, 0.f, 0.f, 0.f, 0.f, 0.f};
#pragma unroll
            for (int mt = 0; mt < 4; ++mt) {
#pragma unroll
                for (int nt = 0; nt < 2; ++nt) {
                    v8f acc = vz;
                    const int row = mt * 16 + l16;
                    const bool rv = row < NN;
                    const int rr = rv ? row : 0;
                    const int col = nt * 16 + l16;
#pragma unroll
                    for (int kk = 0; kk < 14; ++kk) {
                        const int k0 = kk * 4 + half * 2;
                        v2f af;
                        af.x = s_att[rr * 56 + k0];
                        af.y = s_att[rr * 56 + k0 + 1];
                        if (!rv) { af.x = 0.f; af.y = 0.f; }
                        v2f bf;
                        bf.x = s_attri[k0 * HH + col];
                        bf.y = s_attri[(k0 + 1) * HH + col];
                        acc = wmma4(af, bf, acc);
                    }
#pragma unroll
                    for (int v = 0; v < 8; ++v) {
                        const int r = mt * 16 + v + half * 8;
                        float val = acc[v];
                        if (r < NA) s_xA[r * HH + col] = val;
                        else if (r < NN) s_xM[(r - NA) * HH + col] = val;
                    }
                }
            }
        }

        // GRU updates (weights streamed from global as B-fragments; L2-resident)
        gru_block(s_xA, s_hA, NA, 3, g_wihA, g_whhA, s_bihA, s_bhhA, lane);
        gru_block(s_xM, s_hM, NM, 2, g_wihM, g_whhM, s_bihM, s_bhhM, lane);
    }

    // ---------------- epilogue ----------------
    for (int i = lane; i < NA * HH; i += 32)
        d_out[(size_t)b * (NA * HH) + i] = s_hA[i];
    const size_t moff = (size_t)nb * (NA * HH);
    for (int i = lane; i < NM * HH; i += 32)
        d_out[moff + (size_t)b * (NM * HH) + i] = s_hM[i];
}

extern "C" void kernel_launch(void* const* d_in, const int* in_sizes, int n_in,
                              void* d_out, int out_size, void* d_ws, size_t ws_size,
                              hipStream_t stream) {
    const float* X_aqi = (const float*)d_in[0];
    const float* X_meo = (const float*)d_in[1];
    const float* ctx   = (const float*)d_in[2];
    const float* adj   = (const float*)d_in[3];
    const float* adjn  = (const float*)d_in[4];
    const float* e_aid = (const float*)d_in[5];
    const float* e_amo = (const float*)d_in[6];
    const float* e_awd = (const float*)d_in[7];
    const float* e_ahr = (const float*)d_in[8];
    const float* e_mwi = (const float*)d_in[9];
    const float* e_mid = (const float*)d_in[10];
    const float* e_mmo = (const float*)d_in[11];
    const float* e_mwd = (const float*)d_in[12];
    const float* e_mhr = (const float*)d_in[13];
    const float* W_xa  = (const float*)d_in[14];
    const float* W_xm  = (const float*)d_in[15];
    const float* W_ua  = (const float*)d_in[16];
    const float* W_um  = (const float*)d_in[17];
    const float* a_aa  = (const float*)d_in[18];
    const float* a_am  = (const float*)d_in[19];
    const float* a_ma  = (const float*)d_in[20];
    const float* a_mm  = (const float*)d_in[21];
    const float* wihA  = (const float*)d_in[22];
    const float* whhA  = (const float*)d_in[23];
    const float* bihA  = (const float*)d_in[24];
    const float* bhhA  = (const float*)d_in[25];
    const float* wihM  = (const float*)d_in[26];
    const float* whhM  = (const float*)d_in[27];
    const float* bihM  = (const float*)d_in[28];
    const float* bhhM  = (const float*)d_in[29];
    const int* Xae     = (const int*)d_in[30];
    const int* Xme     = (const int*)d_in[31];

    const int nb = in_sizes[0] / (TT * NA * 6);   // batch size (1024)

    chgat_gru_kernel<<<dim3(nb), dim3(32), 0, stream>>>(
        X_aqi, X_meo, ctx, adj, adjn,
        e_aid, e_amo, e_awd, e_ahr, e_mwi, e_mid, e_mmo, e_mwd, e_mhr,
        W_xa, W_xm, W_ua, W_um, a_aa, a_am, a_ma, a_mm,
        wihA, whhA, bihA, bhhA, wihM, whhM, bihM, bhhM,
        Xae, Xme, (float*)d_out, nb);
}